// VQMoEDecoder_11347303596248
// MI455X (gfx1250) — compile-verified
//
#include <hip/hip_runtime.h>
#include <math.h>

#define B_     32
#define N_     1024
#define D_     256
#define KK_    8192
#define TH_    256
#define HEADS_ 8
#define DH_    32
#define FF_    256
#define L_     2
#define BN_SCALE 0.9999950000374997f

typedef float v2f __attribute__((ext_vector_type(2)));
typedef float v8f __attribute__((ext_vector_type(8)));

__device__ inline v8f wmma4(v2f a, v2f b, v8f c) {
  // D = A(16x4) * B(4x16) + C, exact fp32
  return __builtin_amdgcn_wmma_f32_16x16x4_f32(false, a, false, b, (short)0, c, false, false);
}

__device__ inline float softplusf_(float x) {
  if (x > 20.f) return x;
  return log1pf(expf(x));
}

// ---------------- VQ: argmin over codebook, q = E[idx], rowloss ----------------
__global__ __launch_bounds__(256) void vq_kernel(const float* __restrict__ z,
                                                 const float* __restrict__ E,
                                                 float* __restrict__ q,
                                                 float* __restrict__ rowloss) {
  __shared__ float zrow[D_];
  __shared__ float bval[256];
  __shared__ int   bidx[256];
  int b = blockIdx.x, t = threadIdx.x;
  zrow[t] = z[b * D_ + t];
  __syncthreads();
  float best = INFINITY; int bi = 0;
  for (int k = t; k < KK_; k += 256) {
    const float* e = E + (size_t)k * D_;
    float d2 = 0.f;
#pragma unroll 4
    for (int dd = 0; dd < D_; ++dd) { float df = e[dd] - zrow[dd]; d2 = fmaf(df, df, d2); }
    if (d2 < best) { best = d2; bi = k; }
  }
  bval[t] = best; bidx[t] = bi;
  __syncthreads();
  for (int s = 128; s > 0; s >>= 1) {
    if (t < s) {
      if (bval[t + s] < bval[t] || (bval[t + s] == bval[t] && bidx[t + s] < bidx[t])) {
        bval[t] = bval[t + s]; bidx[t] = bidx[t + s];
      }
    }
    __syncthreads();
  }
  int sidx = bidx[0];
  __syncthreads();
  float qv = E[(size_t)sidx * D_ + t];
  q[b * D_ + t] = qv;
  float df = qv - zrow[t];
  bval[t] = df * df;
  __syncthreads();
  for (int s = 128; s > 0; s >>= 1) { if (t < s) bval[t] += bval[t + s]; __syncthreads(); }
  if (t == 0) rowloss[b] = bval[0];
}

__global__ void vq_finalize(const float* __restrict__ rowloss, float* __restrict__ out) {
  if (threadIdx.x == 0 && blockIdx.x == 0) {
    float s = 0.f;
    for (int i = 0; i < B_; ++i) s += rowloss[i];
    out[0] = 1.25f * s / (float)(B_ * D_);   // q_loss + 0.25*e_loss, forward-equal
  }
}

// ---------------- tiny linears (B=32 rows) ----------------
// act: 0=none 1=relu(x*BN_SCALE) 2=relu 3=softplus+1e-4 4=max(softplus,0.1)
__global__ __launch_bounds__(64) void small_linear(const float* __restrict__ X,
                                                   const float* __restrict__ W,
                                                   const float* __restrict__ bias,
                                                   float* __restrict__ Y,
                                                   int K, int O, int act) {
  int b = blockIdx.x;
  int o = blockIdx.y * 64 + threadIdx.x;
  if (o >= O) return;
  const float* x = X + (size_t)b * K;
  const float* w = W + (size_t)o * K;
  float s = 0.f;
#pragma unroll 4
  for (int k = 0; k < K; ++k) s = fmaf(x[k], w[k], s);
  s += bias[o];
  if (act == 1)      s = fmaxf(s * BN_SCALE, 0.f);
  else if (act == 2) s = fmaxf(s, 0.f);
  else if (act == 3) s = softplusf_(s) + 1e-4f;
  else if (act == 4) s = fmaxf(softplusf_(s), 0.1f);
  Y[(size_t)b * O + o] = s;
}

// ---------------- router: hard gumbel-softmax forward value = one-hot ----------------
__global__ void router_kernel(const float* __restrict__ logits, const float* __restrict__ gumbel,
                              float* __restrict__ wts, float* __restrict__ outw) {
  int b = threadIdx.x;
  if (b >= B_) return;
  float l0 = logits[b * 2 + 0] + gumbel[b * 2 + 0];
  float l1 = logits[b * 2 + 1] + gumbel[b * 2 + 1];
  float w0 = (l0 >= l1) ? 1.f : 0.f;   // argmax, first index wins ties
  wts[b * 2 + 0] = w0;  wts[b * 2 + 1] = 1.f - w0;
  outw[b * 2 + 0] = w0; outw[b * 2 + 1] = 1.f - w0;
}

// ---------------- experts -> centered point clouds -> mixed ----------------
__global__ __launch_bounds__(256) void mix_kernel(const float* __restrict__ basis,
                                                  const float* __restrict__ r,
                                                  const float* __restrict__ s,
                                                  const float* __restrict__ wts,
                                                  float* __restrict__ mixed) {
  __shared__ float sL[N_ * 3];
  __shared__ float sA[N_ * 3];
  __shared__ float red[256];
  int b = blockIdx.x, t = threadIdx.x;
  float bas[9];
#pragma unroll
  for (int i = 0; i < 9; ++i) bas[i] = basis[b * 9 + i];
  float sv = s[b];
  float accL[3] = {0.f, 0.f, 0.f}, accA[3] = {0.f, 0.f, 0.f};
  for (int n = t; n < N_; n += 256) {
    // base grid, side=11, t=linspace(0,1,11)
    int i = n / 121, rem = n % 121;
    int j = rem / 11, k = rem % 11;
    float gx = i * 0.1f, gy = j * 0.1f, gz = k * 0.1f;
    float pl[3];
#pragma unroll
    for (int c = 0; c < 3; ++c) pl[c] = gx * bas[c] + gy * bas[3 + c] + gz * bas[6 + c];
    // fibonacci sphere
    float fi = (float)n + 0.5f;
    float phi = 6.2831853071795862f * fi / 1.6180339887498949f;
    float ct = 1.0f - 2.0f * fi / (float)N_;
    float st = sqrtf(fmaxf(1.0f - ct * ct, 0.f));
    float rv = r[b * N_ + n] * sv;
    float pa[3] = { cosf(phi) * st * rv, sinf(phi) * st * rv, ct * rv };
#pragma unroll
    for (int c = 0; c < 3; ++c) {
      sL[n * 3 + c] = pl[c]; sA[n * 3 + c] = pa[c];
      accL[c] += pl[c];      accA[c] += pa[c];
    }
  }
  float mL[3], mA[3];
  for (int c = 0; c < 3; ++c) {
    __syncthreads(); red[t] = accL[c]; __syncthreads();
    for (int sf = 128; sf > 0; sf >>= 1) { if (t < sf) red[t] += red[t + sf]; __syncthreads(); }
    mL[c] = red[0] * (1.f / N_);
    __syncthreads(); red[t] = accA[c]; __syncthreads();
    for (int sf = 128; sf > 0; sf >>= 1) { if (t < sf) red[t] += red[t + sf]; __syncthreads(); }
    mA[c] = red[0] * (1.f / N_);
  }
  float w0 = wts[b * 2 + 0], w1 = wts[b * 2 + 1];
  for (int n = t; n < N_; n += 256) {
#pragma unroll
    for (int c = 0; c < 3; ++c)
      mixed[((size_t)b * N_ + n) * 3 + c] =
          w0 * (sL[n * 3 + c] - mL[c]) + w1 * (sA[n * 3 + c] - mA[c]);
  }
}

// ---------------- h0 = mixed@We^T + be + zproj ----------------
__global__ __launch_bounds__(256) void embed_kernel(const float* __restrict__ mixed,
                                                    const float* __restrict__ We,
                                                    const float* __restrict__ be,
                                                    const float* __restrict__ zproj,
                                                    float* __restrict__ h) {
  size_t idx = blockIdx.x;            // b*N + n
  int b = (int)(idx >> 10);
  int t = threadIdx.x;
  float m0 = mixed[idx * 3 + 0], m1 = mixed[idx * 3 + 1], m2 = mixed[idx * 3 + 2];
  h[idx * TH_ + t] = m0 * We[t * 3 + 0] + m1 * We[t * 3 + 1] + m2 * We[t * 3 + 2]
                   + be[t] + zproj[(size_t)b * TH_ + t];
}

// ------- WMMA fp32 GEMM, 32x32 register tile per wave:
// ------- Y[M,Nout] = X[M,K] * W[Nout,K]^T + bias, optional relu -------
__global__ __launch_bounds__(256) void gemm_wmma(const float* __restrict__ X,
                                                 const float* __restrict__ W,
                                                 const float* __restrict__ bias,
                                                 float* __restrict__ Y,
                                                 int M, int K, int Nout, int act) {
  int wave = blockIdx.x * 8 + (threadIdx.x >> 5);
  int ntn = Nout >> 5;               // 32-wide column tiles
  int tn = (wave % ntn) * 32;
  int tm = (wave / ntn) * 32;
  if (tm >= M) return;               // wave-uniform -> EXEC stays all-ones for WMMA
  int lane = threadIdx.x & 31;
  int half = lane >> 4;
  int l16  = lane & 15;
  // A frag (16x4 f32): lanes 0-15 hold K=k,k+1 of row M=l16; lanes 16-31 hold K=k+2,k+3
  const float* A0 = X + (size_t)(tm + l16) * K + 2 * half;
  const float* A1 = A0 + (size_t)16 * K;
  // B frag (4x16): B[k'][n] = W[n][k']; same per-lane K split, row index = output feature
  const float* B0 = W + (size_t)(tn + l16) * K + 2 * half;
  const float* B1 = B0 + (size_t)16 * K;
  v8f acc00 = {0.f,0.f,0.f,0.f,0.f,0.f,0.f,0.f};
  v8f acc01 = acc00, acc10 = acc00, acc11 = acc00;
#pragma unroll 8
  for (int k = 0; k < K; k += 4) {
    v2f a0 = *(const v2f*)(A0 + k);
    v2f a1 = *(const v2f*)(A1 + k);
    v2f b0 = *(const v2f*)(B0 + k);
    v2f b1 = *(const v2f*)(B1 + k);
    acc00 = wmma4(a0, b0, acc00);
    acc01 = wmma4(a0, b1, acc01);
    acc10 = wmma4(a1, b0, acc10);
    acc11 = wmma4(a1, b1, acc11);
  }
  float bv0 = bias[tn + l16];
  float bv1 = bias[tn + 16 + l16];
  float* Yp = Y + (size_t)tm * Nout + tn;
#pragma unroll
  for (int v = 0; v < 8; ++v) {      // C/D layout: row = v + 8*half, col = l16
    int r0 = v + 8 * half;
    float x00 = acc00[v] + bv0;
    float x01 = acc01[v] + bv1;
    float x10 = acc10[v] + bv0;
    float x11 = acc11[v] + bv1;
    if (act == 1) {
      x00 = fmaxf(x00, 0.f); x01 = fmaxf(x01, 0.f);
      x10 = fmaxf(x10, 0.f); x11 = fmaxf(x11, 0.f);
    }
    Yp[(size_t)r0 * Nout + l16]             = x00;
    Yp[(size_t)r0 * Nout + 16 + l16]        = x01;
    Yp[(size_t)(16 + r0) * Nout + l16]      = x10;
    Yp[(size_t)(16 + r0) * Nout + 16 + l16] = x11;
  }
}

// ---------------- flash attention: one wave per (b, head, 16-query tile) ----------------
__global__ __launch_bounds__(32) void attn_kernel(const float* __restrict__ qkv,
                                                  float* __restrict__ attn) {
  __shared__ __align__(16) float pP[16][18];   // P-tile staging, C-layout -> A-layout
  int blk = blockIdx.x;
  int qt = blk & 63;
  int hd = (blk >> 6) & 7;
  int b  = blk >> 9;
  int lane = threadIdx.x;
  int half = lane >> 4;
  int l16  = lane & 15;
  const int ROW = 3 * TH_;           // 768

  // Q fragments for the whole dh=32 K-loop (8 frags)
  const size_t rowQ = ((size_t)b * N_ + qt * 16 + l16) * ROW + hd * DH_;
  v2f qa[8];
#pragma unroll
  for (int kk = 0; kk < 8; ++kk)
    qa[kk] = *(const v2f*)(qkv + rowQ + kk * 4 + 2 * half);

  v8f o0 = {0.f,0.f,0.f,0.f,0.f,0.f,0.f,0.f};
  v8f o1 = {0.f,0.f,0.f,0.f,0.f,0.f,0.f,0.f};
  float mrow[8], lrow[8];
#pragma unroll
  for (int v = 0; v < 8; ++v) { mrow[v] = -INFINITY; lrow[v] = 0.f; }

  const float sc = 0.17677669529663687f;  // 1/sqrt(32)

  for (int kt = 0; kt < 64; ++kt) {
    // S tile = Q(16x32) * K^T(32x16)
    v8f s = {0.f,0.f,0.f,0.f,0.f,0.f,0.f,0.f};
    const size_t rowK = ((size_t)b * N_ + kt * 16 + l16) * ROW + TH_ + hd * DH_;
#pragma unroll
    for (int kk = 0; kk < 8; ++kk) {
      v2f bb = *(const v2f*)(qkv + rowK + kk * 4 + 2 * half);
      s = wmma4(qa[kk], bb, s);
    }
    // online softmax per row (rows = v + 8*half, cols = 16 lanes of a half-group)
#pragma unroll
    for (int v = 0; v < 8; ++v) {
      float x = s[v] * sc;
      float rmax = x;
      for (int off = 1; off < 16; off <<= 1)
        rmax = fmaxf(rmax, __shfl_xor(rmax, off, 32));
      float mnew = fmaxf(mrow[v], rmax);
      float scale_old = expf(mrow[v] - mnew);
      float p = expf(x - mnew);
      float psum = p;
      for (int off = 1; off < 16; off <<= 1)
        psum += __shfl_xor(psum, off, 32);
      lrow[v] = lrow[v] * scale_old + psum;
      mrow[v] = mnew;
      o0[v] *= scale_old;
      o1[v] *= scale_old;
      pP[v + 8 * half][l16] = p;     // same-wave DS ops stay in order
    }
    // O += P(16x16) * V(16x32)
    const float* Vb = qkv + ((size_t)b * N_ + kt * 16) * ROW + 2 * TH_ + hd * DH_;
#pragma unroll
    for (int kk = 0; kk < 4; ++kk) {
      int krow = kk * 4 + 2 * half;
      v2f pa = *(const v2f*)&pP[l16][krow];
      v2f b0, b1;
      b0.x = Vb[(size_t)krow * ROW + l16];
      b0.y = Vb[(size_t)(krow + 1) * ROW + l16];
      b1.x = Vb[(size_t)krow * ROW + 16 + l16];
      b1.y = Vb[(size_t)(krow + 1) * ROW + 16 + l16];
      o0 = wmma4(pa, b0, o0);
      o1 = wmma4(pa, b1, o1);
    }
  }
  float* Ob = attn + ((size_t)b * N_ + qt * 16) * TH_ + hd * DH_;
#pragma unroll
  for (int v = 0; v < 8; ++v) {
    float inv = 1.0f / lrow[v];
    Ob[(size_t)(v + 8 * half) * TH_ + l16]      = o0[v] * inv;
    Ob[(size_t)(v + 8 * half) * TH_ + 16 + l16] = o1[v] * inv;
  }
}

// ---------------- residual + LayerNorm (in place on h) ----------------
__global__ __launch_bounds__(256) void residual_ln(float* __restrict__ h,
                                                   const float* __restrict__ add,
                                                   const float* __restrict__ g,
                                                   const float* __restrict__ bb) {
  __shared__ float red[256];
  size_t base = (size_t)blockIdx.x * TH_;
  int t = threadIdx.x;
  float x = h[base + t] + add[base + t];
  red[t] = x; __syncthreads();
  for (int s = 128; s > 0; s >>= 1) { if (t < s) red[t] += red[t + s]; __syncthreads(); }
  float mean = red[0] * (1.f / TH_);
  __syncthreads();
  float d = x - mean;
  red[t] = d * d; __syncthreads();
  for (int s = 128; s > 0; s >>= 1) { if (t < s) red[t] += red[t + s]; __syncthreads(); }
  float var = red[0] * (1.f / TH_);
  h[base + t] = d * rsqrtf(var + 1e-5f) * g[t] + bb[t];
}

// ---------------- final = mixed + h@Wo^T + bo ----------------
__global__ __launch_bounds__(256) void final_kernel(const float* __restrict__ h,
                                                    const float* __restrict__ mixed,
                                                    const float* __restrict__ Wo,
                                                    const float* __restrict__ bo,
                                                    float* __restrict__ out) {
  int gid = blockIdx.x * 256 + threadIdx.x;   // < B*N*3
  int idx = gid / 3, c = gid % 3;
  const float* hr = h + (size_t)idx * TH_;
  const float* w = Wo + c * TH_;
  float acc = 0.f;
#pragma unroll 4
  for (int t = 0; t < TH_; ++t) acc = fmaf(hr[t], w[t], acc);
  out[gid] = mixed[gid] + acc + bo[c];
}

extern "C" void kernel_launch(void* const* d_in, const int* in_sizes, int n_in,
                              void* d_out, int out_size, void* d_ws, size_t ws_size,
                              hipStream_t stream) {
  (void)in_sizes; (void)n_in; (void)out_size; (void)ws_size;
  const float* z      = (const float*)d_in[0];
  const float* gumbel = (const float*)d_in[1];
  const float* E      = (const float*)d_in[2];
  const float* Wb1 = (const float*)d_in[3];  const float* bb1 = (const float*)d_in[4];
  const float* Wb2 = (const float*)d_in[5];  const float* bb2 = (const float*)d_in[6];
  const float* Wr1 = (const float*)d_in[7];  const float* br1 = (const float*)d_in[8];
  const float* Wr2 = (const float*)d_in[9];  const float* br2 = (const float*)d_in[10];
  const float* Wr3 = (const float*)d_in[11]; const float* br3 = (const float*)d_in[12];
  const float* Ws1 = (const float*)d_in[13]; const float* bs1 = (const float*)d_in[14];
  const float* Ws2 = (const float*)d_in[15]; const float* bs2 = (const float*)d_in[16];
  const float* Wro1 = (const float*)d_in[17]; const float* bro1 = (const float*)d_in[18];
  const float* Wro2 = (const float*)d_in[19]; const float* bro2 = (const float*)d_in[20];
  const float* We = (const float*)d_in[21]; const float* be = (const float*)d_in[22];
  const float* Wz = (const float*)d_in[23]; const float* bz = (const float*)d_in[24];
  const float* Wqkv = (const float*)d_in[25]; const float* bqkv = (const float*)d_in[26];
  const float* Wmo  = (const float*)d_in[27]; const float* bmo  = (const float*)d_in[28];
  const float* Wf1  = (const float*)d_in[29]; const float* bf1  = (const float*)d_in[30];
  const float* Wf2  = (const float*)d_in[31]; const float* bf2  = (const float*)d_in[32];
  const float* g1n  = (const float*)d_in[33]; const float* b1n  = (const float*)d_in[34];
  const float* g2n  = (const float*)d_in[35]; const float* b2n  = (const float*)d_in[36];
  const float* Wo   = (const float*)d_in[37]; const float* bo   = (const float*)d_in[38];
  float* out = (float*)d_out;

  float* w = (float*)d_ws;
  size_t off = 0;
  auto alloc = [&](size_t n) { float* p = w + off; off += n; return p; };
  float* q       = alloc((size_t)B_ * D_);
  float* rowloss = alloc(B_);
  float* hb      = alloc((size_t)B_ * 256);
  float* basis   = alloc((size_t)B_ * 9);
  float* hr1     = alloc((size_t)B_ * 256);
  float* hr2     = alloc((size_t)B_ * 256);
  float* rbuf    = alloc((size_t)B_ * N_);
  float* hs      = alloc((size_t)B_ * 256);
  float* sbuf    = alloc(B_);
  float* hro     = alloc((size_t)B_ * 128);
  float* logits  = alloc((size_t)B_ * 2);
  float* wts     = alloc((size_t)B_ * 2);
  float* mixed   = alloc((size_t)B_ * N_ * 3);
  float* zproj   = alloc((size_t)B_ * TH_);
  float* h       = alloc((size_t)B_ * N_ * TH_);
  float* qkvb    = alloc((size_t)B_ * N_ * 3 * TH_);
  float* attnb   = alloc((size_t)B_ * N_ * TH_);
  float* tmp     = alloc((size_t)B_ * N_ * TH_);
  float* tmp2    = alloc((size_t)B_ * N_ * TH_);

  // --- VQ ---
  vq_kernel<<<B_, 256, 0, stream>>>(z, E, q, rowloss);
  vq_finalize<<<1, 32, 0, stream>>>(rowloss, out + (size_t)B_ * N_ * 3);

  // --- experts / router (zq forward value == q) ---
  small_linear<<<dim3(B_, 4), 64, 0, stream>>>(q, Wb1, bb1, hb, D_, 256, 1);
  small_linear<<<dim3(B_, 1), 64, 0, stream>>>(hb, Wb2, bb2, basis, 256, 9, 0);
  small_linear<<<dim3(B_, 4), 64, 0, stream>>>(q, Wr1, br1, hr1, D_, 256, 1);
  small_linear<<<dim3(B_, 4), 64, 0, stream>>>(hr1, Wr2, br2, hr2, 256, 256, 1);
  small_linear<<<dim3(B_, 16), 64, 0, stream>>>(hr2, Wr3, br3, rbuf, 256, N_, 3);
  small_linear<<<dim3(B_, 4), 64, 0, stream>>>(q, Ws1, bs1, hs, D_, 256, 1);
  small_linear<<<dim3(B_, 1), 64, 0, stream>>>(hs, Ws2, bs2, sbuf, 256, 1, 4);
  small_linear<<<dim3(B_, 2), 64, 0, stream>>>(q, Wro1, bro1, hro, D_, 128, 2);
  small_linear<<<dim3(B_, 1), 64, 0, stream>>>(hro, Wro2, bro2, logits, 128, 2, 0);
  small_linear<<<dim3(B_, 4), 64, 0, stream>>>(z, Wz, bz, zproj, D_, TH_, 0);
  router_kernel<<<1, 32, 0, stream>>>(logits, gumbel, wts, out + (size_t)B_ * N_ * 3 + 1);
  mix_kernel<<<B_, 256, 0, stream>>>(basis, rbuf, sbuf, wts, mixed);

  // --- transformer refiner ---
  embed_kernel<<<B_ * N_, 256, 0, stream>>>(mixed, We, be, zproj, h);
  const int M = B_ * N_;                         // 32768
  for (int l = 0; l < L_; ++l) {
    gemm_wmma<<<(M / 32) * (3 * TH_ / 32) / 8, 256, 0, stream>>>(
        h, Wqkv + (size_t)l * 3 * TH_ * TH_, bqkv + (size_t)l * 3 * TH_, qkvb, M, TH_, 3 * TH_, 0);
    attn_kernel<<<B_ * HEADS_ * (N_ / 16), 32, 0, stream>>>(qkvb, attnb);
    gemm_wmma<<<(M / 32) * (TH_ / 32) / 8, 256, 0, stream>>>(
        attnb, Wmo + (size_t)l * TH_ * TH_, bmo + (size_t)l * TH_, tmp, M, TH_, TH_, 0);
    residual_ln<<<M, 256, 0, stream>>>(h, tmp, g1n + (size_t)l * TH_, b1n + (size_t)l * TH_);
    gemm_wmma<<<(M / 32) * (FF_ / 32) / 8, 256, 0, stream>>>(
        h, Wf1 + (size_t)l * FF_ * TH_, bf1 + (size_t)l * FF_, tmp2, M, TH_, FF_, 1);
    gemm_wmma<<<(M / 32) * (TH_ / 32) / 8, 256, 0, stream>>>(
        tmp2, Wf2 + (size_t)l * TH_ * FF_, bf2 + (size_t)l * TH_, tmp, M, FF_, TH_, 0);
    residual_ln<<<M, 256, 0, stream>>>(h, tmp, g2n + (size_t)l * TH_, b2n + (size_t)l * TH_);
  }

  final_kernel<<<(B_ * N_ * 3) / 256, 256, 0, stream>>>(h, mixed, Wo, bo, out);
}